// BayesianAtlas_29360396435827
// MI455X (gfx1250) — compile-verified
//
#include <hip/hip_runtime.h>

// Trilinear gather-interpolation of a 3-component field on a 128^3 grid.
// Memory-bound (AI ~0.1 flop/byte): optimize the data-movement path, not math.
//   - points / out are streamed once  -> non-temporal (TH=NT) loads/stores,
//     keeping MI455X's 192MB L2 free for the randomly-gathered vector field.
//   - the two w-taps are adjacent     -> one 8-byte load per (u,v,component)
//     corner: 12 gathers per point instead of 24.
//   - grid-stride loop + global_prefetch_b8 for next iteration's point reads.

#define DGS 128
#define NBP (DGS * DGS * DGS)      // 2^21
#define NBP_SHIFT 21

struct F2 { float x, y; };         // align(4) on purpose: gather offsets have arbitrary parity

__global__ __launch_bounds__(256, 1) void trilerp_kernel(
    const float* __restrict__ vec,   // [bts][3][dgs][dgs][dgs]
    const float* __restrict__ pts,   // [bts][3][dgs][dgs][dgs]
    const int*   __restrict__ dsf_p, // scalar downsampling factor
    float*       __restrict__ out,   // [bts][3][dgs][dgs][dgs]
    int total)                       // bts * NBP
{
    const float invd = 1.0f / (float)(*dsf_p);   // uniform scalar load
    const int stride = (int)(gridDim.x * blockDim.x);

    for (int tid = (int)(blockIdx.x * blockDim.x + threadIdx.x);
         tid < total; tid += stride) {

        // Prefetch next iteration's streamed point data into cache
        // (gfx1250 global_prefetch_b8; speculative, safe).
        int nxt = tid + stride;
        if (nxt < total) {
            int nb = nxt >> NBP_SHIFT;
            int np = nxt & (NBP - 1);
            const float* pp = pts + (size_t)nb * 3u * NBP + np;
            __builtin_prefetch(pp, 0, 1);
            __builtin_prefetch(pp + NBP, 0, 1);
            __builtin_prefetch(pp + 2 * (size_t)NBP, 0, 1);
        }

        const int b = tid >> NBP_SHIFT;
        const int p = tid & (NBP - 1);
        const size_t base = (size_t)b * 3u * NBP;

        // Streamed, single-use reads: non-temporal so they don't evict the field from L2.
        const float px = __builtin_nontemporal_load(pts + base + p);
        const float py = __builtin_nontemporal_load(pts + base + NBP + p);
        const float pz = __builtin_nontemporal_load(pts + base + 2 * (size_t)NBP + p);

        // map to downsampled grid coords: u = (x+1)/dsf - 1
        const float u = (px + 1.0f) * invd - 1.0f;
        const float v = (py + 1.0f) * invd - 1.0f;
        const float w = (pz + 1.0f) * invd - 1.0f;

        float flu = fminf(fmaxf(floorf(u), 0.0f), (float)(DGS - 1));
        float flv = fminf(fmaxf(floorf(v), 0.0f), (float)(DGS - 1));
        float flw = fminf(fmaxf(floorf(w), 0.0f), (float)(DGS - 1));
        const int iul = (int)flu, ivl = (int)flv, iwl = (int)flw;
        const int iuh = min(iul + 1, DGS - 1);
        const int ivh = min(ivl + 1, DGS - 1);
        const int iwh = min(iwl + 1, DGS - 1);
        const float fu = u - flu, gu = flu + 1.0f - u;
        const float fv = v - flv, gv = flv + 1.0f - v;
        const float fw = w - flw, gw = flw + 1.0f - w;

        // Pair-load base along w. When clamped at the top edge (iwh==iwl),
        // shift the pair down one so both taps come from in-bounds memory;
        // a select picks the right lane.
        const bool wclamp = (iwh == iwl);
        const int  wb     = wclamp ? (iwl - 1) : iwl;

        const int r00 = iul * (DGS * DGS) + ivl * DGS + wb;
        const int r01 = iul * (DGS * DGS) + ivh * DGS + wb;
        const int r10 = iuh * (DGS * DGS) + ivl * DGS + wb;
        const int r11 = iuh * (DGS * DGS) + ivh * DGS + wb;
        const float w00 = gu * gv, w01 = gu * fv, w10 = fu * gv, w11 = fu * fv;

#pragma unroll
        for (int c = 0; c < 3; ++c) {
            const float* vc = vec + base + (size_t)c * NBP;
            const F2 f00 = *reinterpret_cast<const F2*>(vc + r00);
            const F2 f01 = *reinterpret_cast<const F2*>(vc + r01);
            const F2 f10 = *reinterpret_cast<const F2*>(vc + r10);
            const F2 f11 = *reinterpret_cast<const F2*>(vc + r11);
            // low-w tap value; when clamped both taps read the same (edge) cell
            const float l00 = wclamp ? f00.y : f00.x;
            const float l01 = wclamp ? f01.y : f01.x;
            const float l10 = wclamp ? f10.y : f10.x;
            const float l11 = wclamp ? f11.y : f11.x;

            const float acc = w00 * (gw * l00 + fw * f00.y)
                            + w01 * (gw * l01 + fw * f01.y)
                            + w10 * (gw * l10 + fw * f10.y)
                            + w11 * (gw * l11 + fw * f11.y);

            // streamed, write-once output: non-temporal store
            __builtin_nontemporal_store(acc, out + base + (size_t)c * NBP + p);
        }
    }
}

extern "C" void kernel_launch(void* const* d_in, const int* in_sizes, int n_in,
                              void* d_out, int out_size, void* d_ws, size_t ws_size,
                              hipStream_t stream) {
    const float* vec   = (const float*)d_in[0];
    const float* pts   = (const float*)d_in[1];
    const int*   dsf_p = (const int*)d_in[2];
    float*       out   = (float*)d_out;

    const int total = in_sizes[0] / 3;   // bts * dgs^3 points
    // Grid-stride: ~4 iterations/thread so the next-iteration prefetch has work to hide.
    const int block = 256;
    int grid = (total + block - 1) / block / 4;
    if (grid < 1) grid = 1;
    if (grid > 65535 * 16) grid = 65535 * 16;

    trilerp_kernel<<<grid, block, 0, stream>>>(vec, pts, dsf_p, out, total);
}